// PDNConvNN_8693013807644
// MI455X (gfx1250) — compile-verified
//
#include <hip/hip_runtime.h>
#include <hip/hip_bf16.h>
#include <math.h>

typedef float v2f __attribute__((ext_vector_type(2)));
typedef float v8f __attribute__((ext_vector_type(8)));

#define NODES 100000
#define HID   128
#define NEDGE 1600000
#define NLAY  5
#define DIN   36
#define HE    32     // 4 * edge_dim
#define EDIM  8
#define OUTF  30     // out_feat * FH

// ---------------------------------------------------------------------------
// 1) Embedding: h = relu(layernorm(x @ emb_w.T + emb_b))
//    One wave (32 lanes) per node; each lane owns 4 of the 128 features.
// ---------------------------------------------------------------------------
__global__ void embed_kernel(const float* __restrict__ x,
                             const float* __restrict__ emb_w,
                             const float* __restrict__ emb_b,
                             const float* __restrict__ ln_g,
                             const float* __restrict__ ln_b,
                             float* __restrict__ h) {
  __shared__ float xs[8][DIN];               // 8 nodes per 256-thread block
  const int t = threadIdx.x;
  const int nodeBase = blockIdx.x * 8;
  for (int i = t; i < 8 * DIN; i += 256) {
    int nn = i / DIN, k = i % DIN;
    xs[nn][k] = x[(size_t)(nodeBase + nn) * DIN + k];
  }
  __syncthreads();

  const int wave = t >> 5, lane = t & 31;
  const int n = nodeBase + wave;

  float v[4];
#pragma unroll
  for (int r = 0; r < 4; ++r) {
    const int j = lane + 32 * r;
    float acc = emb_b[j];
#pragma unroll
    for (int k = 0; k < DIN; ++k) acc = fmaf(xs[wave][k], emb_w[j * DIN + k], acc);
    v[r] = acc;
  }
  // wave32 reduction for mean
  float s = v[0] + v[1] + v[2] + v[3];
#pragma unroll
  for (int off = 16; off > 0; off >>= 1) s += __shfl_xor(s, off, 32);
  const float mu = s * (1.0f / 128.0f);
  float q = 0.0f;
#pragma unroll
  for (int r = 0; r < 4; ++r) { float d = v[r] - mu; q = fmaf(d, d, q); }
#pragma unroll
  for (int off = 16; off > 0; off >>= 1) q += __shfl_xor(q, off, 32);
  const float rs = rsqrtf(q * (1.0f / 128.0f) + 1e-5f);

#pragma unroll
  for (int r = 0; r < 4; ++r) {
    const int j = lane + 32 * r;
    float val = (v[r] - mu) * rs * ln_g[j] + ln_b[j];
    h[(size_t)n * HID + j] = fmaxf(val, 0.0f);
  }
}

// ---------------------------------------------------------------------------
// 2) Edge gates (all 5 layers at once; they do not depend on h).
// ---------------------------------------------------------------------------
__global__ void deg_init_kernel(float* __restrict__ deg, int count) {
  int i = blockIdx.x * blockDim.x + threadIdx.x;
  if (i < count) deg[i] = 1.0f;              // self-loop weight
}

__global__ void edge_gate_kernel(const float* __restrict__ ea,
                                 const long long* __restrict__ ei,
                                 const float* __restrict__ mw1,
                                 const float* __restrict__ mb1,
                                 const float* __restrict__ mw2,
                                 const float* __restrict__ mb2,
                                 float* __restrict__ ew,
                                 float* __restrict__ deg) {
  __shared__ float w1s[HE * EDIM];
  __shared__ float b1s[HE];
  __shared__ float w2s[HE];
  __shared__ float b2s;
  const int l = blockIdx.y;
  const int t = threadIdx.x;
  if (t < HE * EDIM) w1s[t] = mw1[(size_t)l * HE * EDIM + t];
  if (t < HE) { b1s[t] = mb1[l * HE + t]; w2s[t] = mw2[l * HE + t]; }
  if (t == 0) b2s = mb2[l];
  __syncthreads();

  const long long e = (long long)blockIdx.x * 256 + t;
  if (e >= NEDGE) return;
  float a[EDIM];
  const float4* ap = reinterpret_cast<const float4*>(ea + (size_t)e * EDIM);
  float4 a0 = ap[0], a1 = ap[1];
  a[0]=a0.x; a[1]=a0.y; a[2]=a0.z; a[3]=a0.w; a[4]=a1.x; a[5]=a1.y; a[6]=a1.z; a[7]=a1.w;

  float z = b2s;
#pragma unroll 4
  for (int hh = 0; hh < HE; ++hh) {
    float hv = b1s[hh];
#pragma unroll
    for (int k = 0; k < EDIM; ++k) hv = fmaf(a[k], w1s[hh * EDIM + k], hv);
    z = fmaf(fmaxf(hv, 0.0f), w2s[hh], z);
  }
  const float g = 1.0f / (1.0f + __expf(-z));
  ew[(size_t)l * NEDGE + e] = g;
  const int dst = (int)ei[NEDGE + e];
  atomicAdd(&deg[l * NODES + dst], g);
}

__global__ void rsqrt_kernel(float* __restrict__ deg, int count) {
  int i = blockIdx.x * blockDim.x + threadIdx.x;
  if (i < count) { float d = deg[i]; deg[i] = d > 0.0f ? rsqrtf(d) : 0.0f; }
}

// ---------------------------------------------------------------------------
// 3a) hlin = relu(h) @ W.T + b   — exact f32 WMMA (V_WMMA_F32_16X16X4_F32).
//     Block = 256 thr = 8 waves; block owns 16 nodes x 128 cols; wave owns a
//     16x16 tile. h tile staged in LDS with fused ReLU.
// ---------------------------------------------------------------------------
__global__ void lin_wmma_kernel(const float* __restrict__ h,
                                const float* __restrict__ W,
                                const float* __restrict__ bias,
                                float* __restrict__ out) {
  __shared__ float hs[16 * HID];             // 8 KB
  const int t = threadIdx.x;
  const int n0 = blockIdx.x * 16;
  for (int i = t; i < 16 * HID; i += 256)
    hs[i] = fmaxf(h[(size_t)(n0 + (i >> 7)) * HID + (i & 127)], 0.0f);
  __syncthreads();

  const int wave = t >> 5, lane = t & 31;
  const int j0 = wave * 16;
  const int m = lane & 15;                   // A: row index / B: col index
  const int kh = (lane >> 4) * 2;            // K sub-offset within a K=4 step

  v8f c = {};
#pragma unroll 4
  for (int kb = 0; kb < HID; kb += 4) {
    v2f av, bv;
    av.x = hs[m * HID + kb + kh + 0];
    av.y = hs[m * HID + kb + kh + 1];
    bv.x = W[(size_t)(j0 + m) * HID + kb + kh + 0];
    bv.y = W[(size_t)(j0 + m) * HID + kb + kh + 1];
    c = __builtin_amdgcn_wmma_f32_16x16x4_f32(false, av, false, bv,
                                              (short)0, c, false, false);
  }
  const int nhi = lane >> 4;
  const float bj = bias[j0 + m];
#pragma unroll
  for (int r = 0; r < 8; ++r) {
    const int mm = r + 8 * nhi;
    out[(size_t)(n0 + mm) * HID + (j0 + m)] = c[r] + bj;
  }
}

// 3b) acc = hlin * dinv^2 + conv_b   (self-loop contribution + bias)
__global__ void init_acc_kernel(const float* __restrict__ hlin,
                                const float* __restrict__ dinv,
                                const float* __restrict__ cb,
                                float* __restrict__ acc) {
  const int i = blockIdx.x * 256 + threadIdx.x;   // NODES*HID / 256 exact
  const int n = i >> 7;
  const float d = dinv[n];
  acc[i] = fmaf(hlin[i], d * d, cb[i & 127]);
}

// 3c) edge scatter: one wave per edge, lane carries a float4 (128 feats).
__global__ void scatter_kernel(const float* __restrict__ hlin,
                               const long long* __restrict__ ei,
                               const float* __restrict__ ew,
                               const float* __restrict__ dinv,
                               float* __restrict__ acc) {
  const int t = threadIdx.x;
  const int wave = t >> 5, lane = t & 31;
  const long long e = (long long)blockIdx.x * 8 + wave;
  const int src = (int)ei[e];
  const int dst = (int)ei[NEDGE + e];
  const float norm = dinv[src] * ew[e] * dinv[dst];

  const float4 v = reinterpret_cast<const float4*>(hlin + (size_t)src * HID)[lane];
  float* d = acc + (size_t)dst * HID + lane * 4;
  atomicAdd(d + 0, v.x * norm);
  atomicAdd(d + 1, v.y * norm);
  atomicAdd(d + 2, v.z * norm);
  atomicAdd(d + 3, v.w * norm);
}

// ---------------------------------------------------------------------------
// 4) Decoder: out = relu(h) @ dec_w.T + dec_b  (30 cols = 2 WMMA col tiles)
//    Block = 64 thr = 2 waves, 16 nodes per block.
// ---------------------------------------------------------------------------
__global__ void decode_wmma_kernel(const float* __restrict__ h,
                                   const float* __restrict__ dw,
                                   const float* __restrict__ db,
                                   float* __restrict__ out) {
  __shared__ float hs[16 * HID];
  const int t = threadIdx.x;
  const int n0 = blockIdx.x * 16;
  for (int i = t; i < 16 * HID; i += 64)
    hs[i] = fmaxf(h[(size_t)(n0 + (i >> 7)) * HID + (i & 127)], 0.0f);
  __syncthreads();

  const int wave = t >> 5, lane = t & 31;
  const int j0 = wave * 16;
  const int m = lane & 15;
  const int kh = (lane >> 4) * 2;
  const int j = j0 + m;
  const int jc = j < OUTF ? j : OUTF - 1;    // clamp reads for padded cols

  v8f c = {};
#pragma unroll 4
  for (int kb = 0; kb < HID; kb += 4) {
    v2f av, bv;
    av.x = hs[m * HID + kb + kh + 0];
    av.y = hs[m * HID + kb + kh + 1];
    bv.x = dw[(size_t)jc * HID + kb + kh + 0];
    bv.y = dw[(size_t)jc * HID + kb + kh + 1];
    c = __builtin_amdgcn_wmma_f32_16x16x4_f32(false, av, false, bv,
                                              (short)0, c, false, false);
  }
  if (j < OUTF) {
    const int nhi = lane >> 4;
    const float bj = db[j];
#pragma unroll
    for (int r = 0; r < 8; ++r) {
      const int mm = r + 8 * nhi;
      out[(size_t)(n0 + mm) * OUTF + j] = c[r] + bj;
    }
  }
}

// ---------------------------------------------------------------------------
extern "C" void kernel_launch(void* const* d_in, const int* in_sizes, int n_in,
                              void* d_out, int out_size, void* d_ws, size_t ws_size,
                              hipStream_t stream) {
  const float*     x      = (const float*)d_in[0];
  const long long* ei     = (const long long*)d_in[1];
  const float*     ea     = (const float*)d_in[2];
  const float*     emb_w  = (const float*)d_in[3];
  const float*     emb_b  = (const float*)d_in[4];
  const float*     ln_g   = (const float*)d_in[5];
  const float*     ln_b   = (const float*)d_in[6];
  const float*     lin_w  = (const float*)d_in[7];
  const float*     lin_b  = (const float*)d_in[8];
  const float*     mw1    = (const float*)d_in[9];
  const float*     mb1    = (const float*)d_in[10];
  const float*     mw2    = (const float*)d_in[11];
  const float*     mb2    = (const float*)d_in[12];
  const float*     conv_b = (const float*)d_in[13];
  const float*     dec_w  = (const float*)d_in[14];
  const float*     dec_b  = (const float*)d_in[15];
  float*           outp   = (float*)d_out;

  char* ws = (char*)d_ws;
  const size_t nodeBytes = (size_t)NODES * HID * sizeof(float);   // 51.2 MB
  float* bufA = (float*)(ws);                                     // h / acc
  float* bufB = (float*)(ws + nodeBytes);                         // hlin
  float* ewb  = (float*)(ws + 2 * nodeBytes);                     // 5*E gates
  float* deg  = (float*)(ws + 2 * nodeBytes + (size_t)NLAY * NEDGE * sizeof(float));

  // 1) embedding + layernorm + relu
  embed_kernel<<<NODES / 8, 256, 0, stream>>>(x, emb_w, emb_b, ln_g, ln_b, bufA);

  // 2) edge gates + degrees for all layers
  deg_init_kernel<<<(NLAY * NODES + 255) / 256, 256, 0, stream>>>(deg, NLAY * NODES);
  edge_gate_kernel<<<dim3(NEDGE / 256, NLAY), 256, 0, stream>>>(ea, ei, mw1, mb1,
                                                                mw2, mb2, ewb, deg);
  rsqrt_kernel<<<(NLAY * NODES + 255) / 256, 256, 0, stream>>>(deg, NLAY * NODES);

  // 3) five PDNConv layers (ReLU fused into next layer's loads)
  for (int l = 0; l < NLAY; ++l) {
    lin_wmma_kernel<<<NODES / 16, 256, 0, stream>>>(bufA,
                                                    lin_w + (size_t)l * HID * HID,
                                                    lin_b + (size_t)l * HID, bufB);
    init_acc_kernel<<<NODES * HID / 256, 256, 0, stream>>>(bufB, deg + (size_t)l * NODES,
                                                           conv_b + (size_t)l * HID, bufA);
    scatter_kernel<<<NEDGE / 8, 256, 0, stream>>>(bufB, ei,
                                                  ewb + (size_t)l * NEDGE,
                                                  deg + (size_t)l * NODES, bufA);
  }

  // 4) decoder
  decode_wmma_kernel<<<NODES / 16, 64, 0, stream>>>(bufA, dec_w, dec_b, outp);
}